// PolicyGCN_111669150307
// MI455X (gfx1250) — compile-verified
//
#include <hip/hip_runtime.h>
#include <hip/hip_bf16.h>
#include <math.h>

#define N_NODES 20000
#define N_EDGES 160000
#define N_GRAPHS 16
#define FIN 5
#define H 1000
#define NEG_INF_F (-1.0e30f)

typedef __attribute__((ext_vector_type(2))) float v2f;
typedef __attribute__((ext_vector_type(8))) float v8f;

// ---------------------------------------------------------------- init / degree
__global__ void init_kernel(float* __restrict__ deg, int* __restrict__ count) {
  int i = blockIdx.x * blockDim.x + threadIdx.x;
  if (i < N_NODES) { deg[i] = 2.0f; count[i] = 0; }   // self-loop weight 2.0 (improved=True)
}

__global__ void deg_kernel(const int* __restrict__ col, const float* __restrict__ ew,
                           float* __restrict__ deg, int* __restrict__ count) {
  int e = blockIdx.x * blockDim.x + threadIdx.x;
  if (e < N_EDGES) {
    int c = col[e];
    atomicAdd(&deg[c], ew[e]);
    atomicAdd(&count[c], 1);
  }
}

__global__ void dinv_kernel(const float* __restrict__ deg, float* __restrict__ dinv,
                            float* __restrict__ selfnorm) {
  int i = blockIdx.x * blockDim.x + threadIdx.x;
  if (i < N_NODES) {
    float d = deg[i];
    float di = (d > 0.0f) ? (1.0f / sqrtf(d)) : 0.0f;
    dinv[i] = di;
    selfnorm[i] = 2.0f * di * di;   // dinv * 2.0 * dinv  (self loop)
  }
}

// ------------------------------------------------------- one-block prefix scan
__global__ void scan_kernel(const int* __restrict__ count, int* __restrict__ rowptr,
                            int* __restrict__ fillpos) {
  __shared__ int sdata[1024];
  __shared__ int srun;
  int t = threadIdx.x;
  if (t == 0) srun = 0;
  __syncthreads();
  for (int base = 0; base < N_NODES; base += 1024) {
    int i = base + t;
    int v = (i < N_NODES) ? count[i] : 0;
    sdata[t] = v;
    __syncthreads();
    for (int off = 1; off < 1024; off <<= 1) {
      int tv = (t >= off) ? sdata[t - off] : 0;
      __syncthreads();
      sdata[t] += tv;
      __syncthreads();
    }
    int run = srun;
    int excl = sdata[t] - v;
    if (i < N_NODES) { rowptr[i] = run + excl; fillpos[i] = run + excl; }
    __syncthreads();
    if (t == 1023) srun = run + sdata[1023];
    __syncthreads();
  }
  if (t == 0) rowptr[N_NODES] = srun;
}

// ------------------------------------------------- scatter edges into CSR(dst)
__global__ void scatter_kernel(const int* __restrict__ row, const int* __restrict__ col,
                               const float* __restrict__ ew, const float* __restrict__ dinv,
                               int* __restrict__ fillpos, int* __restrict__ csr_src,
                               float* __restrict__ csr_w) {
  int e = blockIdx.x * blockDim.x + threadIdx.x;
  if (e < N_EDGES) {
    int r = row[e], c = col[e];
    float nrm = dinv[r] * ew[e] * dinv[c];
    int p = atomicAdd(&fillpos[c], 1);
    csr_src[p] = r;
    csr_w[p] = nrm;
  }
}

// ------------------------------------------- aggregate x (width 5), no atomics
__global__ void aggx_kernel(const float* __restrict__ x, const int* __restrict__ rowptr,
                            const int* __restrict__ csr_src, const float* __restrict__ csr_w,
                            const float* __restrict__ selfnorm, float* __restrict__ aggx) {
  int i = blockIdx.x * blockDim.x + threadIdx.x;
  if (i >= N_NODES) return;
  float sn = selfnorm[i];
  float a0 = sn * x[i * FIN + 0], a1 = sn * x[i * FIN + 1], a2 = sn * x[i * FIN + 2];
  float a3 = sn * x[i * FIN + 3], a4 = sn * x[i * FIN + 4];
  int p0 = rowptr[i], p1 = rowptr[i + 1];
  for (int p = p0; p < p1; ++p) {
    int s = csr_src[p];
    float wv = csr_w[p];
    const float* xs = x + (size_t)s * FIN;
    a0 += wv * xs[0]; a1 += wv * xs[1]; a2 += wv * xs[2]; a3 += wv * xs[3]; a4 += wv * xs[4];
  }
  float* o = aggx + (size_t)i * FIN;
  o[0] = a0; o[1] = a1; o[2] = a2; o[3] = a3; o[4] = a4;
}

// -------------------------------------------------- tiny GEMM1: [N,5]x[5,1000]
__global__ __launch_bounds__(256) void gemm1_relu_kernel(const float* __restrict__ aggx,
                                                         const float* __restrict__ W1,
                                                         const float* __restrict__ b1,
                                                         float* __restrict__ h1) {
  int node = blockIdx.x;
  float a[FIN];
#pragma unroll
  for (int k = 0; k < FIN; ++k) a[k] = aggx[(size_t)node * FIN + k];
  for (int f = threadIdx.x; f < H; f += 256) {
    float acc = b1[f];
#pragma unroll
    for (int k = 0; k < FIN; ++k) acc += a[k] * W1[k * H + f];
    h1[(size_t)node * H + f] = fmaxf(acc, 0.0f);
  }
}

// --------------------------------------- aggregate h1 (width 1000), no atomics
__global__ __launch_bounds__(256) void agg_wide_kernel(const float* __restrict__ h,
                                                       const int* __restrict__ rowptr,
                                                       const int* __restrict__ csr_src,
                                                       const float* __restrict__ csr_w,
                                                       const float* __restrict__ selfnorm,
                                                       float* __restrict__ outw) {
  int node = blockIdx.x;
  int t = threadIdx.x;
  float sn = selfnorm[node];
  const float* hi = h + (size_t)node * H;
  float a0 = sn * hi[t];
  float a1 = sn * hi[t + 256];
  float a2 = sn * hi[t + 512];
  float a3 = (t + 768 < H) ? sn * hi[t + 768] : 0.0f;
  int p0 = rowptr[node], p1 = rowptr[node + 1];
  for (int p = p0; p < p1; ++p) {
    int s = csr_src[p];
    float wv = csr_w[p];
    const float* hs = h + (size_t)s * H;
    a0 += wv * hs[t];
    a1 += wv * hs[t + 256];
    a2 += wv * hs[t + 512];
    if (t + 768 < H) a3 += wv * hs[t + 768];
  }
  float* o = outw + (size_t)node * H;
  o[t] = a0; o[t + 256] = a1; o[t + 512] = a2;
  if (t + 768 < H) o[t + 768] = a3;
}

// ------------------------------- GEMM2: [20000,1000]x[1000,1000] fp32 via WMMA
// Block: 256 thr / 8 waves, tile 128x64, K-chunk 16, register double buffering.
// B staged in LDS in fragment order so each B-frag is one contiguous 2-dword
// LDS load into an even-aligned VGPR pair (no repack movs); pitch 129 puts the
// two lane-halves on disjoint (even/odd) bank sets.
#define BM 128
#define BN 64
#define BK 16
#define BPITCH 129
__global__ __launch_bounds__(256) void gemm2_wmma_kernel(const float* __restrict__ A,
                                                         const float* __restrict__ B,
                                                         const float* __restrict__ bias,
                                                         float* __restrict__ C) {
  __shared__ float As[BM][BK + 1];        // pad -> conflict-free A-frag reads
  __shared__ float Bsw[8 * BPITCH];       // fragment-ordered B chunk
  const int M = N_NODES, Nn = H, K = H;
  const int bm = blockIdx.x * BM;
  const int bn = blockIdx.y * BN;
  const int tid = threadIdx.x;
  const int w = tid >> 5;        // wave 0..7 -> 16-row strip of the M tile
  const int lane = tid & 31;
  const int lhalf = lane >> 4;   // 0 / 1
  const int l16 = lane & 15;

  // staging indices (fixed per thread)
  const int ar0 = tid >> 2, ac = (tid & 3) << 2;        // A: two rows, 4 cols
  const int ar1 = (tid + 256) >> 2;
  const int bkrow = tid >> 4, bc = (tid & 15) << 2;     // B: 1 krow, 4 cols

  v8f acc[4] = {};
  float4 pa0, pa1, pb;

  // prefetch chunk 0
  {
    float4 z = make_float4(0.f, 0.f, 0.f, 0.f);
    int gm0 = bm + ar0, gm1 = bm + ar1;
    pa0 = (gm0 < M) ? *(const float4*)(A + (size_t)gm0 * K + ac) : z;
    pa1 = (gm1 < M) ? *(const float4*)(A + (size_t)gm1 * K + ac) : z;
    int gn = bn + bc;
    pb = (gn < Nn) ? *(const float4*)(B + (size_t)bkrow * Nn + gn) : z;
  }

  for (int k0 = 0; k0 < K; k0 += BK) {
    // commit staged registers to LDS
    As[ar0][ac + 0] = pa0.x; As[ar0][ac + 1] = pa0.y;
    As[ar0][ac + 2] = pa0.z; As[ar0][ac + 3] = pa0.w;
    As[ar1][ac + 0] = pa1.x; As[ar1][ac + 1] = pa1.y;
    As[ar1][ac + 2] = pa1.z; As[ar1][ac + 3] = pa1.w;
    {
      // krow r = 4*kk + h + 2*second  ->  Bsw[(2*kk+h)*BPITCH + 2*n + second]
      int kk = bkrow >> 2, rem = bkrow & 3;
      int h = rem & 1, second = rem >> 1;
      float* dst = &Bsw[(kk * 2 + h) * BPITCH + second];
      dst[2 * (bc + 0)] = pb.x; dst[2 * (bc + 1)] = pb.y;
      dst[2 * (bc + 2)] = pb.z; dst[2 * (bc + 3)] = pb.w;
    }
    __syncthreads();

    // prefetch next chunk while computing this one
    int kn = k0 + BK;
    if (kn < K) {
      float4 z = make_float4(0.f, 0.f, 0.f, 0.f);
      int gm0 = bm + ar0, gm1 = bm + ar1;
      int gk = kn + ac;
      bool kv = (gk < K);
      pa0 = (gm0 < M && kv) ? *(const float4*)(A + (size_t)gm0 * K + gk) : z;
      pa1 = (gm1 < M && kv) ? *(const float4*)(A + (size_t)gm1 * K + gk) : z;
      int gkb = kn + bkrow, gn = bn + bc;
      pb = (gkb < K && gn < Nn) ? *(const float4*)(B + (size_t)gkb * Nn + gn) : z;
    }

#pragma unroll
    for (int kk = 0; kk < 4; ++kk) {
      const int kb = kk * 4;
      // A-frag 16x4: lanes 0-15 hold K={kb,kb+1}, lanes 16-31 K={kb+2,kb+3}
      const int arow = (w << 4) + l16;
      v2f a;
      a.x = As[arow][kb + (lhalf << 1)];
      a.y = As[arow][kb + (lhalf << 1) + 1];
      const int brow = (kk * 2 + lhalf) * BPITCH;
#pragma unroll
      for (int nt = 0; nt < 4; ++nt) {
        const int bi = brow + 2 * ((nt << 4) + l16);
        v2f b;
        b.x = Bsw[bi];       // B[kb+lhalf][n]
        b.y = Bsw[bi + 1];   // B[kb+2+lhalf][n]
        acc[nt] = __builtin_amdgcn_wmma_f32_16x16x4_f32(
            false, a, false, b, (short)0, acc[nt], false, false);
      }
    }
    __syncthreads();
  }

  // epilogue: bias + relu.  C/D layout: VGPR r -> M = r + 8*lhalf, N = l16
#pragma unroll
  for (int nt = 0; nt < 4; ++nt) {
#pragma unroll
    for (int r = 0; r < 8; ++r) {
      int gm = bm + (w << 4) + r + (lhalf << 3);
      int gn = bn + (nt << 4) + l16;
      if (gm < M && gn < Nn)
        C[(size_t)gm * Nn + gn] = fmaxf(acc[nt][r] + bias[gn], 0.0f);
    }
  }
}

// ------------------------------------------- logits: wave-per-node dot with Wf
__global__ __launch_bounds__(256) void logits_kernel(const float* __restrict__ h2,
                                                     const float* __restrict__ Wf,
                                                     const float* __restrict__ bf,
                                                     float* __restrict__ logits) {
  int wave = threadIdx.x >> 5;
  int lane = threadIdx.x & 31;
  int node = blockIdx.x * 8 + wave;
  if (node >= N_NODES) return;
  const float* hr = h2 + (size_t)node * H;
  float acc = 0.0f;
  for (int f = lane; f < H; f += 32) acc += hr[f] * Wf[f];
#pragma unroll
  for (int off = 16; off > 0; off >>= 1) acc += __shfl_xor(acc, off, 32);
  if (lane == 0) logits[node] = acc + bf[0];
}

// ------------------------------------------------------- masked segment softmax
__global__ __launch_bounds__(256) void seg_max_kernel(const float* __restrict__ logits,
                                                      const unsigned char* __restrict__ mask,
                                                      const int* __restrict__ batch,
                                                      float* __restrict__ mx) {
  int g = blockIdx.x;
  __shared__ float red[256];
  float m = -3.0e38f;
  for (int i = threadIdx.x; i < N_NODES; i += 256)
    if (batch[i] == g) m = fmaxf(m, mask[i] ? logits[i] : NEG_INF_F);
  red[threadIdx.x] = m;
  __syncthreads();
  for (int s = 128; s > 0; s >>= 1) {
    if (threadIdx.x < s) red[threadIdx.x] = fmaxf(red[threadIdx.x], red[threadIdx.x + s]);
    __syncthreads();
  }
  if (threadIdx.x == 0) mx[g] = red[0];
}

__global__ __launch_bounds__(256) void seg_expsum_kernel(const float* __restrict__ logits,
                                                         const unsigned char* __restrict__ mask,
                                                         const int* __restrict__ batch,
                                                         const float* __restrict__ mx,
                                                         float* __restrict__ ex,
                                                         float* __restrict__ ssum) {
  int g = blockIdx.x;
  float mg = mx[g];
  __shared__ float red[256];
  float s = 0.0f;
  for (int i = threadIdx.x; i < N_NODES; i += 256) {
    if (batch[i] == g) {
      float e = mask[i] ? expf(logits[i] - mg) : 0.0f;
      ex[i] = e;
      s += e;
    }
  }
  red[threadIdx.x] = s;
  __syncthreads();
  for (int o = 128; o > 0; o >>= 1) {
    if (threadIdx.x < o) red[threadIdx.x] += red[threadIdx.x + o];
    __syncthreads();
  }
  if (threadIdx.x == 0) ssum[g] = red[0];
}

__global__ void softmax_out_kernel(const float* __restrict__ ex, const float* __restrict__ ssum,
                                   const int* __restrict__ batch, float* __restrict__ out) {
  int i = blockIdx.x * blockDim.x + threadIdx.x;
  if (i < N_NODES) out[i] = ex[i] / fmaxf(ssum[batch[i]], 1e-16f);
}

// ------------------------------------------------------------------------------
extern "C" void kernel_launch(void* const* d_in, const int* in_sizes, int n_in,
                              void* d_out, int out_size, void* d_ws, size_t ws_size,
                              hipStream_t stream) {
  (void)in_sizes; (void)n_in; (void)out_size; (void)ws_size;
  const float* x  = (const float*)d_in[0];
  const float* ew = (const float*)d_in[1];
  const float* W1 = (const float*)d_in[2];
  const float* b1 = (const float*)d_in[3];
  const float* W2 = (const float*)d_in[4];
  const float* b2 = (const float*)d_in[5];
  const float* Wf = (const float*)d_in[6];
  const float* bf = (const float*)d_in[7];
  const int*   ei = (const int*)d_in[8];
  const unsigned char* mask = (const unsigned char*)d_in[9];  // bool = 1B/elem
  const int* batch = (const int*)d_in[10];
  float* out = (float*)d_out;
  const int* row = ei;            // edge_index[0]
  const int* col = ei + N_EDGES;  // edge_index[1]

  // workspace carve-up (256B aligned)
  char* base = (char*)d_ws;
  size_t off = 0;
  auto carve = [&](size_t bytes) -> void* {
    off = (off + 255) & ~(size_t)255;
    void* p = base + off;
    off += bytes;
    return p;
  };
  float* deg      = (float*)carve(N_NODES * 4);
  float* dinv     = (float*)carve(N_NODES * 4);
  float* selfnorm = (float*)carve(N_NODES * 4);
  int*   count    = (int*)carve(N_NODES * 4);
  int*   rowptr   = (int*)carve((N_NODES + 1) * 4);
  int*   fillpos  = (int*)carve(N_NODES * 4);
  int*   csr_src  = (int*)carve(N_EDGES * 4);
  float* csr_w    = (float*)carve(N_EDGES * 4);
  float* aggx     = (float*)carve((size_t)N_NODES * FIN * 4);
  float* logitsv  = (float*)carve(N_NODES * 4);
  float* exv      = (float*)carve(N_NODES * 4);
  float* mx       = (float*)carve(N_GRAPHS * 4);
  float* ssum     = (float*)carve(N_GRAPHS * 4);
  float* bufA     = (float*)carve((size_t)N_NODES * H * 4);  // h1, later h2
  float* bufB     = (float*)carve((size_t)N_NODES * H * 4);  // aggregated h1

  const int TPB = 256;
  int gN = (N_NODES + TPB - 1) / TPB;
  int gE = (N_EDGES + TPB - 1) / TPB;

  init_kernel<<<gN, TPB, 0, stream>>>(deg, count);
  deg_kernel<<<gE, TPB, 0, stream>>>(col, ew, deg, count);
  dinv_kernel<<<gN, TPB, 0, stream>>>(deg, dinv, selfnorm);
  scan_kernel<<<1, 1024, 0, stream>>>(count, rowptr, fillpos);
  scatter_kernel<<<gE, TPB, 0, stream>>>(row, col, ew, dinv, fillpos, csr_src, csr_w);

  // layer 1: aggregate(x) then transform (A and W commute)
  aggx_kernel<<<gN, TPB, 0, stream>>>(x, rowptr, csr_src, csr_w, selfnorm, aggx);
  gemm1_relu_kernel<<<N_NODES, TPB, 0, stream>>>(aggx, W1, b1, bufA);

  // layer 2: aggregate(h1) then WMMA transform
  agg_wide_kernel<<<N_NODES, TPB, 0, stream>>>(bufA, rowptr, csr_src, csr_w, selfnorm, bufB);
  dim3 g2((N_NODES + BM - 1) / BM, (H + BN - 1) / BN);
  gemm2_wmma_kernel<<<g2, TPB, 0, stream>>>(bufB, W2, b2, bufA);  // bufA now = h2

  logits_kernel<<<N_NODES / 8, TPB, 0, stream>>>(bufA, Wf, bf, logitsv);
  seg_max_kernel<<<N_GRAPHS, TPB, 0, stream>>>(logitsv, mask, batch, mx);
  seg_expsum_kernel<<<N_GRAPHS, TPB, 0, stream>>>(logitsv, mask, batch, mx, exv, ssum);
  softmax_out_kernel<<<gN, TPB, 0, stream>>>(exv, ssum, batch, out);
}